// YOLOv3Loss_38362647888013
// MI455X (gfx1250) — compile-verified
//
#include <hip/hip_runtime.h>
#include <math.h>

typedef __attribute__((ext_vector_type(2))) float v2f;
typedef __attribute__((ext_vector_type(8))) float v8f;

#define B_   16
#define A_   3
#define T_   50
#define C_   80
#define CH_  85

#define N13  (16*3*13*13)   /* 8112   */
#define N26  (16*3*26*26)   /* 32448  */
#define N52  (16*3*52*52)   /* 129792 */

#define NB13 32             /* ceil(8112/256)   */
#define NB26 127            /* ceil(32448/256)  */
#define NB52 507            /* 129792/256       */
#define PB13 0
#define PB26 (PB13+NB13)    /* 32  */
#define PB52 (PB26+NB26)    /* 159 */
#define NROWS (PB52+NB52)   /* 666 */

#define EPSF 1e-12f

__device__ __forceinline__ float bce_t1(float p) {   // -log(p), p clipped
    p = fminf(fmaxf(p, EPSF), 1.0f - EPSF);
    return -logf(p);
}
__device__ __forceinline__ float bce_t0(float p) {   // -log1p(-p), p clipped
    p = fminf(fmaxf(p, EPSF), 1.0f - EPSF);
    return -log1pf(-p);
}

// ---------------------------------------------------------------------------
// Cell code: bit0 = no-obj mask (1 = contributes to noobj); bits[31:1] =
// (recordIndex+1) if cell is a positive, else 0.
// ---------------------------------------------------------------------------
__global__ void init_cells_kernel(int* __restrict__ cells, int n) {
    int i = blockIdx.x * 256 + threadIdx.x;
    if (i < n) cells[i] = 1;
}

// One thread per (scale, batch): serial over the 50 targets of that batch so
// scatter collisions resolve deterministically (last target wins, matching a
// sequential .at[].set; min/max-style bit clears are order-invariant anyway).
__global__ void build_targets_kernel(const float* __restrict__ tgt,
                                     const float* __restrict__ a13,
                                     const float* __restrict__ a26,
                                     const float* __restrict__ a52,
                                     int* __restrict__ cells13,
                                     int* __restrict__ cells26,
                                     int* __restrict__ cells52,
                                     float* __restrict__ recs,
                                     int* __restrict__ recCls) {
    int tid = threadIdx.x;
    if (tid >= 48) return;
    int s = tid / 16;
    int b = tid % 16;
    int g = (s == 0) ? 13 : ((s == 1) ? 26 : 52);
    const float* anc = (s == 0) ? a13 : ((s == 1) ? a26 : a52);
    int* cells = (s == 0) ? cells13 : ((s == 1) ? cells26 : cells52);
    float gf = (float)g;

    for (int t = 0; t < T_; ++t) {
        const float* tp = tgt + ((size_t)b * T_ + t) * 5;
        float x1 = tp[0], y1 = tp[1], x2 = tp[2], y2 = tp[3];
        int   cls = (int)tp[4];
        float bx = 0.5f * (x1 + x2) * gf;
        float by = 0.5f * (y1 + y2) * gf;
        float bw = (x2 - x1) * gf;
        float bh = (y2 - y1) * gf;

        float ious[3];
        float bestIoU = -1.0f;
        int best = 0;
        for (int a = 0; a < 3; ++a) {
            float aw = anc[a * 2 + 0], ah = anc[a * 2 + 1];
            float inter = fminf(aw, bw) * fminf(ah, bh);
            float uni   = 1e-8f + aw * ah + bw * bh - inter;
            float iou   = inter / uni;
            ious[a] = iou;
            if (iou > bestIoU) { bestIoU = iou; best = a; }  // first max wins
        }
        int gi = (int)bx; gi = min(max(gi, 0), g - 1);
        int gj = (int)by; gj = min(max(gj, 0), g - 1);

        // nm &= (iou <= thres) at this target's cell, for every anchor
        for (int a = 0; a < 3; ++a) {
            if (ious[a] > 0.5f) {
                int c = ((b * 3 + a) * g + gj) * g + gi;
                cells[c] &= ~1;
            }
        }
        // Positive record (last target writing a cell wins)
        int r  = b * T_ + t;            // record within scale
        int gr = s * (B_ * T_) + r;     // global record
        recs[gr * 4 + 0] = bx - floorf(bx);
        recs[gr * 4 + 1] = by - floorf(by);
        recs[gr * 4 + 2] = logf(bw / anc[best * 2 + 0]);
        recs[gr * 4 + 3] = logf(bh / anc[best * 2 + 1]);
        recCls[gr] = cls;
        int c = ((b * 3 + best) * g + gj) * g + gi;
        cells[c] = ((r + 1) << 1);      // positive; nm bit forced 0
    }
}

// ---------------------------------------------------------------------------
// Dense pass: one thread per cell. Components:
// 0:n_pos 1:n_neg 2:lx 3:ly 4:lw 5:lh 6:obj 7:noobj 8:cls  (9..15 = 0 pad)
// ---------------------------------------------------------------------------
__global__ void dense_loss_kernel(const float* __restrict__ out,
                                  const int*   __restrict__ cells,
                                  const float* __restrict__ recs,
                                  const int*   __restrict__ recCls,
                                  float* __restrict__ partials, int N) {
    int tid = threadIdx.x;
    int i = blockIdx.x * 256 + tid;
    float v[9];
#pragma unroll
    for (int k = 0; k < 9; ++k) v[k] = 0.f;

    if (i < N) {
        int code = cells[i];
        const float* cp = out + (size_t)i * CH_;
        float p4 = cp[4];
        float nmv = (float)(code & 1);
        v[1] = nmv;
        v[7] = nmv * bce_t0(p4);
        if (code >> 1) {
            int r = (code >> 1) - 1;
            const float* rp = recs + (size_t)r * 4;
            int cls = recCls[r];
            v[0] = 1.0f;
            float dx = cp[0] - rp[0]; v[2] = dx * dx;
            float dy = cp[1] - rp[1]; v[3] = dy * dy;
            float dw = cp[2] - rp[2]; v[4] = dw * dw;
            float dh = cp[3] - rp[3]; v[5] = dh * dh;
            v[6] = bce_t1(p4);
            float cs = 0.f;
            for (int c = 0; c < C_; ++c) {
                float p = cp[5 + c];
                cs += (c == cls) ? bce_t1(p) : bce_t0(p);
            }
            v[8] = cs;
        }
    }

    __shared__ float sm[256];
    float* row = partials + (size_t)blockIdx.x * 16;
#pragma unroll
    for (int k = 0; k < 9; ++k) {
        sm[tid] = v[k];
        __syncthreads();
        for (int off = 128; off > 0; off >>= 1) {
            if (tid < off) sm[tid] += sm[tid + off];
            __syncthreads();
        }
        if (tid == 0) row[k] = sm[0];
        __syncthreads();
    }
    if (tid == 0) {
#pragma unroll
        for (int k = 9; k < 16; ++k) row[k] = 0.f;
    }
}

// ---------------------------------------------------------------------------
// Finisher (1 wave32): WMMA-based partial-row reduction.
// A = ones(16x4)  =>  D[m][n] = sum_k B[k][n], so the result is invariant to
// the K-row <-> lane-half mapping; only N = lane%16 matters. Accumulate 4
// partial rows per V_WMMA_F32_16X16X4_F32 into the f32 accumulator, then
// read component sums from accumulator VGPR0 (M=0 row, N=lane).
// ---------------------------------------------------------------------------
__global__ void final_reduce_kernel(const float* __restrict__ partials,
                                    float* __restrict__ out) {
    const int nrows[3] = {NB13, NB26, NB52};
    const int base[3]  = {PB13, PB26, PB52};
    int lane = threadIdx.x;
    int half = lane >> 4;
    int col  = lane & 15;
    __shared__ float comp[16];

    v2f a; a[0] = 1.0f; a[1] = 1.0f;   // all-ones A matrix
    float total = 0.f;

    for (int s = 0; s < 3; ++s) {
        const float* P = partials + (size_t)base[s] * 16;
        int n = nrows[s];
        v8f acc = {};
        for (int r0 = 0; r0 < n; r0 += 4) {
            int ra = r0 + half;        // rows r0, r0+1 across lane halves
            int rb = r0 + 2 + half;    // rows r0+2, r0+3
            v2f bb;
            bb[0] = (ra < n) ? P[(size_t)ra * 16 + col] : 0.f;
            bb[1] = (rb < n) ? P[(size_t)rb * 16 + col] : 0.f;
            acc = __builtin_amdgcn_wmma_f32_16x16x4_f32(
                false, a, false, bb, (short)0, acc, false, false);
        }
        if (lane < 16) comp[lane] = acc[0];   // M=0 row: component sums
        __syncthreads();
        if (lane == 0) {
            float np = comp[0], nn = comp[1];
            total += (comp[2] + comp[3] + comp[4] + comp[5] + comp[6]) / np
                   + comp[7] / nn * 100.0f
                   + comp[8] / (np * (float)C_);
        }
        __syncthreads();
    }
    if (lane == 0) out[0] = total;
}

// ---------------------------------------------------------------------------
extern "C" void kernel_launch(void* const* d_in, const int* in_sizes, int n_in,
                              void* d_out, int out_size, void* d_ws, size_t ws_size,
                              hipStream_t stream) {
    (void)in_sizes; (void)n_in; (void)out_size; (void)ws_size;
    const float* out13 = (const float*)d_in[0];
    const float* out26 = (const float*)d_in[1];
    const float* out52 = (const float*)d_in[2];
    const float* tgt   = (const float*)d_in[3];
    const float* a13   = (const float*)d_in[4];
    const float* a26   = (const float*)d_in[5];
    const float* a52   = (const float*)d_in[6];

    // Workspace layout (all 4B-aligned, ~770 KB total)
    float* partials = (float*)d_ws;                       // NROWS * 16 floats
    int*   cells13  = (int*)(partials + (size_t)NROWS * 16);
    int*   cells26  = cells13 + N13;
    int*   cells52  = cells26 + N26;
    float* recs     = (float*)(cells52 + N52);            // 3*800 recs * 4 floats
    int*   recCls   = (int*)(recs + (size_t)3 * B_ * T_ * 4);

    const int totalCells = N13 + N26 + N52;
    init_cells_kernel<<<(totalCells + 255) / 256, 256, 0, stream>>>(cells13, totalCells);

    build_targets_kernel<<<1, 64, 0, stream>>>(tgt, a13, a26, a52,
                                               cells13, cells26, cells52,
                                               recs, recCls);

    dense_loss_kernel<<<NB13, 256, 0, stream>>>(out13, cells13,
        recs,                         recCls,             partials + (size_t)PB13 * 16, N13);
    dense_loss_kernel<<<NB26, 256, 0, stream>>>(out26, cells26,
        recs + (size_t)(B_*T_) * 4,   recCls + B_*T_,     partials + (size_t)PB26 * 16, N26);
    dense_loss_kernel<<<NB52, 256, 0, stream>>>(out52, cells52,
        recs + (size_t)(2*B_*T_) * 4, recCls + 2*B_*T_,   partials + (size_t)PB52 * 16, N52);

    final_reduce_kernel<<<1, 32, 0, stream>>>(partials, (float*)d_out);
}